// QConv2d_12128987644268
// MI455X (gfx1250) — compile-verified
//
#include <hip/hip_runtime.h>
#include <stdint.h>

typedef __attribute__((ext_vector_type(2))) float v2f;
typedef __attribute__((ext_vector_type(8))) float v8f;
typedef __attribute__((ext_vector_type(4))) unsigned int u32x4;
typedef __attribute__((ext_vector_type(4))) int i32x4;
typedef __attribute__((ext_vector_type(8))) int i32x8;

#define CIN    256
#define COUT   256
#define HW_    56
#define PIX    3136           // 56*56
#define NBATCH 16
#define KTOT   2304           // CIN*3*3
#define BM     64             // channel tile
#define BN     128            // pixel tile
#define BK     16             // k tile (4 wmma k-steps)
#define NIT    (KTOT / BK)    // 144
#define AS_STRIDE  18         // floats per A row in LDS (even -> aligned b64 frags,
                              //  conflict-free: 18*row mod 64 distinct for 16 rows)
#define AS_FLOATS  (BM * AS_STRIDE)        // 1152 floats per buffer
#define BS2_STRIDE 144        // float2 per k-pair row (2*144 mod 64 == 32 ->
                              //  lane-halves hit disjoint bank ranges)

// ---- Tensor Data Mover: load one 16(k) x 64(row) f32 weight tile into LDS ----
// D# per CDNA5 ISA 8.3/8.4: count=1, type=2(image), data_size=4B,
// tile_dim0=16, tile_dim1=64, tensor_dim0_stride=KTOT,
// pad_enable: every 16 DWORDs insert 2 DWORDs -> LDS row stride 18 floats.
__device__ __forceinline__ void tdm_load_weights(const float* gptr, unsigned lds_byte)
{
    unsigned long long ga = (unsigned long long)(uintptr_t)gptr;   // byte address
    u32x4 g0;
    g0[0] = 1u;                                            // count=1 (valid), user mode
    g0[1] = lds_byte;                                      // lds_addr (bytes)
    g0[2] = (unsigned)(ga & 0xFFFFFFFFu);                  // global_addr[31:0]
    g0[3] = (unsigned)((ga >> 32) & 0x01FFFFFFu) | (2u << 30);  // addr[56:32] | type=2
    i32x8 g1;
    g1[0] = (2 << 16) | (1 << 20) | (3 << 22) | (1 << 25); // 4B | pad_en | every16DW | +2DW
    g1[1] = (int)((KTOT & 0xFFFF) << 16);                  // tensor_dim0[15:0]
    g1[2] = (int)((KTOT >> 16) | ((COUT & 0xFFFF) << 16)); // dim0 hi | tensor_dim1 lo
    g1[3] = (int)(BK << 16);                               // dim1 hi=0 | tile_dim0=16
    g1[4] = (int)BM;                                       // tile_dim1=64 | tile_dim2=0
    g1[5] = (int)KTOT;                                     // tensor_dim0_stride lo
    g1[6] = 0;
    g1[7] = 0;
    i32x4 z4 = {0, 0, 0, 0};
#if defined(__clang_major__) && __clang_major__ >= 23
    i32x8 z8 = {0, 0, 0, 0, 0, 0, 0, 0};
    __builtin_amdgcn_tensor_load_to_lds(g0, g1, z4, z4, z8, 0);
#else
    __builtin_amdgcn_tensor_load_to_lds(g0, g1, z4, z4, 0);
#endif
}

__global__ __launch_bounds__(256)
void qconv2d_wmma_f32(const float* __restrict__ x,
                      const float* __restrict__ w,
                      const float* __restrict__ bias,
                      float* __restrict__ out)
{
    __shared__ float As[2][AS_FLOATS];          // 2*1152*4B = 9216B (TDM double buffer)
    __shared__ float Bs[(BK / 2) * BS2_STRIDE * 2];  // 8*144*2*4B = 9216B, pair-interleaved

    const int tid  = threadIdx.x;
    const int wave = tid >> 5;
    const int lane = tid & 31;
    const int half = lane >> 4;
    const int ml   = lane & 15;

    const int bm = blockIdx.x;       // 0..3    channel tile
    const int bn = blockIdx.y;       // 0..391  pixel tile

    const int wm = wave >> 1;        // 0..3 : 16-channel subrow
    const int wn = wave & 1;         // 0..1 : 64-pixel half

    v8f acc[4] = {};

    // ---- hoisted per-thread im2col pixel decode (independent of kt) ----
    int kk_i[8], pl_i[8], xbase_i[8], h_i[8], w_i[8];
    #pragma unroll
    for (int i = 0; i < 8; ++i) {
        int idx = tid + i * 256;
        int kk  = idx >> 7;          // k-row within tile: 0..15
        int pl  = idx & 127;         // pixel within tile
        int p   = bn * BN + pl;
        int n   = p / PIX;
        int phw = p - n * PIX;
        int h   = phw / HW_;
        int ww  = phw - h * HW_;
        kk_i[i] = kk; pl_i[i] = pl;
        h_i[i] = h;  w_i[i] = ww;
        xbase_i[i] = n * (CIN * PIX);            // batch base (c,ih,iw added per kt)
    }

    const float* wtile_base = w + (size_t)(bm * BM) * KTOT;
    const unsigned as_lds   = (unsigned)(uintptr_t)&As[0][0];
    const unsigned as_bytes = AS_FLOATS * 4;

    // prologue: kick off TDM for the first weight tile
    if (tid < 32)
        tdm_load_weights(wtile_base, as_lds);

    for (int it = 0; it < NIT; ++it) {
        const int kt  = it * BK;
        const int cur = it & 1;

        if (tid < 32)
            __builtin_amdgcn_s_wait_tensorcnt(0);   // As[cur] DMA complete
        __syncthreads();                             // prev compute done; Bs & As[cur^1] free

        if ((tid < 32) && (it + 1 < NIT))
            tdm_load_weights(wtile_base + (kt + BK), as_lds + (cur ^ 1) * as_bytes);

        // ---- stage B (im2col of x): 16 k-rows x 128 pixels, pair-interleaved ----
        #pragma unroll
        for (int i = 0; i < 8; ++i) {
            int k  = kt + kk_i[i];
            int c  = k / 9;
            int rs = k - c * 9;
            int r  = rs / 3;
            int s  = rs - r * 3;
            int ih = h_i[i] + r - 1;
            int iw = w_i[i] + s - 1;
            float v = 0.0f;
            if ((unsigned)ih < (unsigned)HW_ && (unsigned)iw < (unsigned)HW_)
                v = x[xbase_i[i] + c * PIX + ih * HW_ + iw];
            int kk = kk_i[i];
            Bs[((kk >> 1) * BS2_STRIDE + pl_i[i]) * 2 + (kk & 1)] = v;
        }
        __syncthreads();

        // ---- compute: 4 k-steps (K=4) x 4 N-subtiles, all frags = one b64 ----
        const float* Ab = &As[cur][0];
        #pragma unroll
        for (int kk4 = 0; kk4 < BK; kk4 += 4) {
            const int ka = kk4 + 2 * half;           // even
            v2f a = *(const v2f*)&Ab[(wm * 16 + ml) * AS_STRIDE + ka];
            #pragma unroll
            for (int j = 0; j < 4; ++j) {
                int col = wn * 64 + j * 16 + ml;
                v2f b = *(const v2f*)&Bs[((ka >> 1) * BS2_STRIDE + col) * 2];
                acc[j] = __builtin_amdgcn_wmma_f32_16x16x4_f32(
                    false, a, false, b, (short)0, acc[j], false, false);
            }
        }
    }

    // ---- epilogue: C/D layout -> NCHW, +bias, coalesced along pixels ----
    const int o_base = bm * BM + wm * 16 + half * 8;
    float bv[8];
    #pragma unroll
    for (int v = 0; v < 8; ++v) bv[v] = bias[o_base + v];

    #pragma unroll
    for (int j = 0; j < 4; ++j) {
        int pcol = bn * BN + wn * 64 + j * 16 + ml;
        int n    = pcol / PIX;
        int phw  = pcol - n * PIX;
        #pragma unroll
        for (int v = 0; v < 8; ++v) {
            int o = o_base + v;
            out[((long)n * COUT + o) * PIX + phw] = acc[j][v] + bv[v];
        }
    }
}

extern "C" void kernel_launch(void* const* d_in, const int* in_sizes, int n_in,
                              void* d_out, int out_size, void* d_ws, size_t ws_size,
                              hipStream_t stream) {
    const float* x    = (const float*)d_in[0];   // (16,256,56,56)
    const float* wgt  = (const float*)d_in[1];   // (256,256,3,3)
    const float* bias = (const float*)d_in[2];   // (256,)
    float* out = (float*)d_out;                  // (16,256,56,56)

    dim3 grid(COUT / BM, (NBATCH * PIX) / BN);   // (4, 392)
    dim3 block(256);
    qconv2d_wmma_f32<<<grid, block, 0, stream>>>(x, wgt, bias, out);
}